// TraditionalSP_20624432955541
// MI455X (gfx1250) — compile-verified
//
#include <hip/hip_runtime.h>

// Problem constants (from reference)
#define MELS  128
#define NKEYS 88
#define KPAD  96          // pad 88 -> 96 = 6 WMMA M-tiles
#define TLEN  16384
#define NB    16
#define TW    64          // time-columns per workgroup
#define BSTR  72          // LDS mel-tile row stride (floats): 2*72 % 64 == 16 -> conflict-free B frags
#define SSTR  97          // LDS S-tile column stride (floats): 97 % 64 = 33, coprime -> conflict-free scans

#define AS1 __attribute__((address_space(1)))
#define AS3 __attribute__((address_space(3)))

typedef __attribute__((ext_vector_type(2))) float v2f;
typedef __attribute__((ext_vector_type(8))) float v8f;
typedef int v4i __attribute__((vector_size(16)));   // 128-bit payload type for async LDS loads

// Build the 96x128 gather/harmonic matrix G from key_bins.
// G[k,kb] += 1;  G[k,2kb] += 1 if kb<64;  G[k,3kb] += 1 if kb<43;  rows >= 88 are zero.
__global__ void build_g_kernel(const int* __restrict__ key_bins, float* __restrict__ G) {
    const int k = threadIdx.x;
    if (k >= KPAD) return;
    const int kb = (k < NKEYS) ? key_bins[k] : -1;
    for (int m = 0; m < MELS; ++m) {
        float g = 0.0f;
        if (kb >= 0) {
            if (m == kb)                 g += 1.0f;
            if (kb < 64 && m == 2 * kb)  g += 1.0f;
            if (kb < 43 && m == 3 * kb)  g += 1.0f;
        }
        G[k * MELS + m] = g;
    }
}

__global__ __launch_bounds__(128)
void hps_key_kernel(const float* __restrict__ mel,
                    const float* __restrict__ G,
                    float* __restrict__ out) {
    __shared__ float sTile[MELS * BSTR];   // 36864 B; reused for S^T after the GEMM

    const int tid = threadIdx.x;
    const int b  = blockIdx.y;
    const int t0 = blockIdx.x * TW;

    // ---- Stage 128x64 f32 mel tile into LDS (one row per thread) ----
    // CDNA5 async global->LDS path: no VGPR round-trip, tracked by ASYNCcnt.
    {
        const int row = tid;  // 128 threads == 128 rows
        const float* src = mel + ((size_t)b * MELS + row) * TLEN + t0;
        float* dst = &sTile[row * BSTR];
#if __has_builtin(__builtin_amdgcn_global_load_async_to_lds_b128)
        #pragma unroll
        for (int i = 0; i < TW / 4; ++i) {
            __builtin_amdgcn_global_load_async_to_lds_b128(
                (AS1 v4i*)(uintptr_t)(src + 4 * i),
                (AS3 v4i*)(uintptr_t)(dst + 4 * i),
                /*imm offset=*/0, /*cpol=*/0);
        }
#if __has_builtin(__builtin_amdgcn_s_wait_asynccnt)
        __builtin_amdgcn_s_wait_asynccnt(0);
#else
        asm volatile("s_wait_asynccnt 0x0" ::: "memory");
#endif
#else
        const float4* s4 = reinterpret_cast<const float4*>(src);
        float4*       d4 = reinterpret_cast<float4*>(dst);
        #pragma unroll
        for (int i = 0; i < TW / 4; ++i) d4[i] = s4[i];
#endif
    }
    __syncthreads();

    // ---- GEMM: S(96 x 64) = G(96 x 128) x melTile(128 x 64) via v_wmma_f32_16x16x4_f32 ----
    const int wave    = tid >> 5;
    const int lane    = tid & 31;
    const int hf      = lane >> 4;   // half-wave select
    const int ln      = lane & 15;
    const int colBase = wave * 16;   // each wave owns 16 columns

    v8f acc[6];
    #pragma unroll
    for (int mt = 0; mt < 6; ++mt) {
        #pragma unroll
        for (int r = 0; r < 8; ++r) acc[mt][r] = 0.0f;
    }

    for (int kk = 0; kk < MELS; kk += 4) {
        // B fragment (4x16, K x N): VGPR0 = K rows {kk, kk+2}, VGPR1 = {kk+1, kk+3}
        v2f bfrag;
        bfrag[0] = sTile[(kk     + 2 * hf) * BSTR + colBase + ln];
        bfrag[1] = sTile[(kk + 1 + 2 * hf) * BSTR + colBase + ln];
        #pragma unroll
        for (int mt = 0; mt < 6; ++mt) {
            // A fragment (16x4): lane ln holds row mt*16+ln, K pair {kk+2h, kk+2h+1} -> one b64 load
            const float2 ag = *reinterpret_cast<const float2*>(
                &G[(mt * 16 + ln) * MELS + kk + 2 * hf]);
            v2f afrag; afrag[0] = ag.x; afrag[1] = ag.y;
#if __has_builtin(__builtin_amdgcn_wmma_f32_16x16x4_f32)
            acc[mt] = __builtin_amdgcn_wmma_f32_16x16x4_f32(
                /*neg_a=*/false, afrag, /*neg_b=*/false, bfrag,
                /*c_mod=*/(short)0, acc[mt], /*reuse_a=*/false, /*reuse_b=*/false);
#else
            acc[mt][0] += afrag[0] * bfrag[0] + afrag[1] * bfrag[1];
#endif
        }
    }
    __syncthreads();   // mel tile fully consumed; overlay S^T in the same LDS

    // ---- Transpose-store S into LDS: sS[col * 97 + k]  (D layout: VGPR r -> M = r + 8*hf) ----
    float* sS = sTile;
    #pragma unroll
    for (int mt = 0; mt < 6; ++mt) {
        #pragma unroll
        for (int r = 0; r < 8; ++r) {
            const int rowk = mt * 16 + 8 * hf + r;
            sS[(colBase + ln) * SSTR + rowk] = acc[mt][r];
        }
    }
    __syncthreads();

    // ---- Selection: probs[k] = 1 iff #{j: s[j] > s[k]} <= 13 (== 85th pct test, exact) ----
    const int col   = tid & (TW - 1);   // 64 columns
    const int khalf = tid >> 6;         // two threads per column split the 88 keys 44/44
    const float* scol = &sS[col * SSTR];
    float* out1 = out + (size_t)NB * NKEYS * TLEN;   // duplicate output (tuple of two)

    #pragma unroll
    for (int chunk = 0; chunk < 2; ++chunk) {
        const int k0 = khalf * 44 + chunk * 22;
        float v[22]; int cnt[22];
        #pragma unroll
        for (int i = 0; i < 22; ++i) { v[i] = scol[k0 + i]; cnt[i] = 0; }
        for (int j = 0; j < NKEYS; ++j) {
            const float sj = scol[j];
            #pragma unroll
            for (int i = 0; i < 22; ++i) cnt[i] += (sj > v[i]) ? 1 : 0;
        }
        #pragma unroll
        for (int i = 0; i < 22; ++i) {
            const float p = (cnt[i] <= 13) ? 1.0f : 0.0f;
            const size_t off = ((size_t)(b * NKEYS + k0 + i)) * TLEN + (size_t)(t0 + col);
            out[off]  = p;   // coalesced: lanes hold consecutive t
            out1[off] = p;
        }
    }
}

extern "C" void kernel_launch(void* const* d_in, const int* in_sizes, int n_in,
                              void* d_out, int out_size, void* d_ws, size_t ws_size,
                              hipStream_t stream) {
    const float* mel      = (const float*)d_in[0];   // (16, 128, 16384) f32
    const int*   key_bins = (const int*)d_in[1];     // (88,) i32
    float*       G        = (float*)d_ws;            // 96*128 f32 = 48 KiB scratch
    float*       out      = (float*)d_out;           // 2 * 16*88*16384 f32

    build_g_kernel<<<dim3(1), dim3(128), 0, stream>>>(key_bins, G);
    hps_key_kernel<<<dim3(TLEN / TW, NB), dim3(128), 0, stream>>>(mel, G, out);
}